// RoiPoolingConv_56195352101291
// MI455X (gfx1250) — compile-verified
//
#include <hip/hip_runtime.h>

#define PS   7
#define NCH  8
#define DIM  256
#define NROI 64

// gfx1250 async global->LDS path (ASYNCcnt-tracked). Guarded so the file
// compiles on both host pass and toolchains lacking the builtin.
#if defined(__HIP_DEVICE_COMPILE__) && \
    __has_builtin(__builtin_amdgcn_global_load_async_to_lds_b64) && \
    __has_builtin(__builtin_amdgcn_s_wait_asynccnt)
#define USE_ASYNC_LDS 1
#endif

__device__ __forceinline__ float4 lerp4(float4 a, float4 b, float f) {
    const float g = 1.0f - f;
    float4 r;
    r.x = a.x * g + b.x * f;
    r.y = a.y * g + b.y * f;
    r.z = a.z * g + b.z * f;
    r.w = a.w * g + b.w * f;
    return r;
}

// Matches reference axis_coords: src = j * (size/7); i0 = floor(src) (always
// <= size-1 since src < size); frac from unclamped i0; i1 = min(i0+1, size-1).
__device__ __forceinline__ void axis_coords(int start, int size, int j,
                                            int& i0, int& i1, float& frac) {
    const float src = (float)j * ((float)size / (float)PS);
    const float fl  = floorf(src);
    frac = src - fl;
    const int a = (int)fl;
    int b = a + 1;
    const int cap = size - 1;
    if (b > cap) b = cap;
    i0 = start + a;
    i1 = start + b;
}

__global__ __launch_bounds__(64) void roi_pool3d_kernel(
    const float* __restrict__ img,   // (256,256,256,8) f32
    const int*   __restrict__ rois,  // (64,6) i32: x,y,z,w,h,d
    float*       __restrict__ out)   // (64,7,7,7,8) f32
{
    const int px  = blockIdx.x;   // 0..6  output x
    const int roi = blockIdx.y;   // 0..63

    __shared__ __align__(16) int sroi[8];
#ifdef USE_ASYNC_LDS
    // Builtin signature (from clang diagnostic): param0 is
    // 'int __vector(2) AS(1)*' (global src), so: (gsrc, lds_dst, imm off, imm cpol).
    // The imm offset is added to BOTH addresses per the ISA (GV mode).
    typedef int v2i __attribute__((__vector_size__(8)));
    typedef __attribute__((address_space(1))) v2i* gp_t;
    typedef __attribute__((address_space(3))) v2i* lp_t;
    if (threadIdx.x == 0) {
        gp_t g = (gp_t)(const_cast<int*>(rois) + roi * 6);  // 24B row, 8B aligned
        lp_t l = (lp_t)sroi;
        __builtin_amdgcn_global_load_async_to_lds_b64(g, l, 0,  0);
        __builtin_amdgcn_global_load_async_to_lds_b64(g, l, 8,  0);
        __builtin_amdgcn_global_load_async_to_lds_b64(g, l, 16, 0);
    }
    __builtin_amdgcn_s_wait_asynccnt(0);
    __syncthreads();
#else
    if (threadIdx.x < 6) sroi[threadIdx.x] = rois[roi * 6 + threadIdx.x];
    __syncthreads();
#endif

    const int tid = threadIdx.x;
    if (tid >= PS * PS) return;            // lanes 49..63 idle (after barrier)
    const int py = tid / PS;
    const int pz = tid - py * PS;

    int xi0, xi1, yi0, yi1, zi0, zi1;
    float fx, fy, fz;
    axis_coords(sroi[0], sroi[3], px, xi0, xi1, fx);
    axis_coords(sroi[1], sroi[4], py, yi0, yi1, fy);
    axis_coords(sroi[2], sroi[5], pz, zi0, zi1, fz);

    const size_t SX = (size_t)DIM * DIM * NCH;   // x stride (floats)
    const size_t SY = (size_t)DIM * NCH;         // y stride (floats)

    const size_t ox0 = (size_t)xi0 * SX, ox1 = (size_t)xi1 * SX;
    const size_t oy0 = (size_t)yi0 * SY, oy1 = (size_t)yi1 * SY;
    const size_t oz0 = (size_t)zi0 * NCH, oz1 = (size_t)zi1 * NCH;

    // 8 corner base pointers; 8 channels = 2 x float4 (b128) each, 32B aligned.
    const float4* p000 = (const float4*)(img + ox0 + oy0 + oz0);
    const float4* p001 = (const float4*)(img + ox0 + oy0 + oz1);
    const float4* p010 = (const float4*)(img + ox0 + oy1 + oz0);
    const float4* p011 = (const float4*)(img + ox0 + oy1 + oz1);
    const float4* p100 = (const float4*)(img + ox1 + oy0 + oz0);
    const float4* p101 = (const float4*)(img + ox1 + oy0 + oz1);
    const float4* p110 = (const float4*)(img + ox1 + oy1 + oz0);
    const float4* p111 = (const float4*)(img + ox1 + oy1 + oz1);

    float4* op = (float4*)(out +
        ((size_t)roi * (PS * PS * PS) + (size_t)px * (PS * PS) + (size_t)tid) * NCH);

#pragma unroll
    for (int half = 0; half < 2; ++half) {
        const float4 v000 = p000[half];
        const float4 v001 = p001[half];
        const float4 v010 = p010[half];
        const float4 v011 = p011[half];
        const float4 v100 = p100[half];
        const float4 v101 = p101[half];
        const float4 v110 = p110[half];
        const float4 v111 = p111[half];
        // Reference order: z-lerp, then y, then x.
        const float4 c00 = lerp4(v000, v001, fz);
        const float4 c01 = lerp4(v010, v011, fz);
        const float4 c10 = lerp4(v100, v101, fz);
        const float4 c11 = lerp4(v110, v111, fz);
        const float4 c0  = lerp4(c00, c01, fy);
        const float4 c1  = lerp4(c10, c11, fy);
        op[half] = lerp4(c0, c1, fx);
    }
}

extern "C" void kernel_launch(void* const* d_in, const int* in_sizes, int n_in,
                              void* d_out, int out_size, void* d_ws, size_t ws_size,
                              hipStream_t stream) {
    (void)in_sizes; (void)n_in; (void)out_size; (void)d_ws; (void)ws_size;
    const float* img  = (const float*)d_in[0];
    const int*   rois = (const int*)d_in[1];
    // d_in[2] = pool_size (==7, fixed at compile time to match reference POOL)
    float* out = (float*)d_out;

    dim3 grid(PS, NROI);     // 448 blocks: one (roi, px) slab each
    dim3 block(64);          // 2 wave32 per block; 49 active lanes
    roi_pool3d_kernel<<<grid, block, 0, stream>>>(img, rois, out);
}